// MN4_80444737454118
// MI455X (gfx1250) — compile-verified
//
#include <hip/hip_runtime.h>
#include <hip/hip_bf16.h>
#include <stdint.h>

typedef __attribute__((ext_vector_type(2))) float v2f;
typedef __attribute__((ext_vector_type(8))) float v8f;

#define B_   4
#define C_   64
#define HW   25
#define NW   5
#define KS   5
#define Q_   75
#define U_   100
#define MS   125      // K_SHOT*h*w
#define MU   2500     // u*h*w
#define MQ   25
#define MTOT 2625
#define SUPC 640      // 625 support cols padded to 40 tiles
#define NCAP 2640     // 2625 padded to 165 tiles
#define EPSN 1e-12f

// ---------------- init ----------------
__global__ void k_init(float* out, int* Lp) {
    if (threadIdx.x == 0) { out[0] = 0.0f; *Lp = 0; }
}

// ---------------- normalization ----------------
__global__ void k_norm_unl(const float* __restrict__ unl, float* __restrict__ unlN) {
    int idx = blockIdx.x * blockDim.x + threadIdx.x;
    if (idx >= B_ * MU) return;
    int b = idx / MU, m = idx % MU;
    int u = m / HW, hw = m % HW;
    const float* src = unl + ((size_t)(b * U_ + u) * C_) * HW + hw;
    float ss = 0.f;
    for (int c = 0; c < C_; ++c) { float v = src[(size_t)c * HW]; ss += v * v; }
    float inv = 1.0f / fmaxf(sqrtf(ss), EPSN);
    for (int c = 0; c < C_; ++c)
        unlN[((size_t)b * C_ + c) * MU + m] = src[(size_t)c * HW] * inv;
}

__global__ void k_norm_sup(const float* __restrict__ sup, float* __restrict__ supN) {
    int idx = blockIdx.x * blockDim.x + threadIdx.x;
    if (idx >= B_ * SUPC) return;
    int b = idx / SUPC, col = idx % SUPC;
    if (col >= NW * MS) {
        for (int c = 0; c < C_; ++c) supN[((size_t)b * C_ + c) * SUPC + col] = 0.f;
        return;
    }
    int w = col / MS, n = col % MS;
    int ks = n / HW, hw = n % HW;
    const float* src = sup + ((size_t)((b * NW + w) * KS + ks) * C_) * HW + hw;
    float ss = 0.f;
    for (int c = 0; c < C_; ++c) { float v = src[(size_t)c * HW]; ss += v * v; }
    float inv = 1.0f / fmaxf(sqrtf(ss), EPSN);
    for (int c = 0; c < C_; ++c)
        supN[((size_t)b * C_ + c) * SUPC + col] = src[(size_t)c * HW] * inv;
}

__global__ void k_norm_q(const float* __restrict__ qx, float* __restrict__ qN) {
    int idx = blockIdx.x * blockDim.x + threadIdx.x;
    if (idx >= B_ * Q_ * 32) return;
    int bq = idx / 32, mq = idx % 32;
    float* dst = qN + (size_t)bq * C_ * 32 + mq;
    if (mq >= MQ) {
        for (int c = 0; c < C_; ++c) dst[(size_t)c * 32] = 0.f;
        return;
    }
    const float* src = qx + ((size_t)bq * C_) * HW + mq;
    float ss = 0.f;
    for (int c = 0; c < C_; ++c) { float v = src[(size_t)c * HW]; ss += v * v; }
    float inv = 1.0f / fmaxf(sqrtf(ss), EPSN);
    for (int c = 0; c < C_; ++c) dst[(size_t)c * 32] = src[(size_t)c * HW] * inv;
}

// ---------------- GEMM1: pool[b][m][col] = unlN^T @ supN (WMMA f32 16x16x4) ----------------
__global__ void __launch_bounds__(32) k_gemm1(const float* __restrict__ unlN,
                                              const float* __restrict__ supN,
                                              float* __restrict__ pool) {
    int nt = blockIdx.x, mt = blockIdx.y, b = blockIdx.z;
    int lane = threadIdx.x, hi = lane >> 4, lo = lane & 15;
    int mg = mt * 16 + lo; if (mg > MU - 1) mg = MU - 1;   // clamp pad rows
    int ng = nt * 16 + lo;
    const float* Ab = unlN + (size_t)b * C_ * MU + mg;
    const float* Bb = supN + (size_t)b * C_ * SUPC + ng;
    v8f acc = {};
    for (int k0 = 0; k0 < C_; k0 += 4) {
        int ka = k0 + 2 * hi;
        v2f a, bv;
        a[0]  = Ab[(size_t)ka * MU];        a[1]  = Ab[(size_t)(ka + 1) * MU];
        bv[0] = Bb[(size_t)ka * SUPC];      bv[1] = Bb[(size_t)(ka + 1) * SUPC];
        acc = __builtin_amdgcn_wmma_f32_16x16x4_f32(false, a, false, bv,
                                                    (short)0, acc, false, false);
    }
    for (int r = 0; r < 8; ++r) {
        int row = mt * 16 + r + 8 * hi;
        int col = nt * 16 + lo;
        if (row < MU) pool[((size_t)b * MU + row) * SUPC + col] = acc[r];
    }
}

// ---------------- per-unlabeled argmaxes ----------------
__global__ void k_unearest(const float* __restrict__ pool, int* __restrict__ u_nearest,
                           int* __restrict__ cls) {
    int idx = blockIdx.x * blockDim.x + threadIdx.x;
    if (idx >= B_ * MU) return;
    int b = idx / MU, m = idx % MU;
    const float* row = pool + ((size_t)b * MU + m) * SUPC;
    float best = -3.0e38f; int bi = 0;
    float wbest[NW]; int cbest = 0; float cmax = -3.0e38f;
    for (int w = 0; w < NW; ++w) {
        float wm = -3.0e38f;
        for (int n = 0; n < MS; ++n) {
            float v = row[w * MS + n];
            if (v > best) { best = v; bi = w * MS + n; }
            if (v > wm) wm = v;
        }
        wbest[w] = wm;
        if (wm > cmax) { cmax = wm; cbest = w; }
    }
    (void)wbest;
    u_nearest[idx] = bi;
    cls[idx] = cbest;
}

// ---------------- per-support-column argmax over m ----------------
__global__ void __launch_bounds__(256) k_snearest(const float* __restrict__ pool,
                                                  int* __restrict__ s_nearest) {
    __shared__ float sv[256];
    __shared__ int   si[256];
    int b = blockIdx.x / (NW * MS), col = blockIdx.x % (NW * MS);
    int tid = threadIdx.x;
    float bv = -3.0e38f; int bi = 0x7fffffff;
    for (int m = tid; m < MU; m += 256) {
        float v = pool[((size_t)b * MU + m) * SUPC + col];
        if (v > bv) { bv = v; bi = m; }  // m increasing -> first-occurrence locally
    }
    sv[tid] = bv; si[tid] = bi;
    __syncthreads();
    for (int s = 128; s > 0; s >>= 1) {
        if (tid < s) {
            if (sv[tid + s] > sv[tid] ||
                (sv[tid + s] == sv[tid] && si[tid + s] < si[tid])) {
                sv[tid] = sv[tid + s]; si[tid] = si[tid + s];
            }
        }
        __syncthreads();
    }
    if (tid == 0) s_nearest[b * (NW * MS) + col] = si[0];
}

// ---------------- mutual-NN + compaction + counts + L ----------------
__global__ void k_compact(const int* __restrict__ u_nearest, const int* __restrict__ s_nearest,
                          const int* __restrict__ cls, int* __restrict__ compidx,
                          int* __restrict__ counts, int* __restrict__ Lp) {
    int b = threadIdx.x;
    if (b >= B_) return;
    int cnt[NW] = {0, 0, 0, 0, 0};
    for (int m = 0; m < MU; ++m) {
        int un = u_nearest[b * MU + m];
        if (s_nearest[b * (NW * MS) + un] == m) {
            int w = cls[b * MU + m];
            compidx[((size_t)(b * NW + w)) * MU + cnt[w]] = m;
            cnt[w]++;
        }
    }
    for (int w = 0; w < NW; ++w) {
        counts[b * NW + w] = cnt[w];
        atomicMax(Lp, cnt[w]);
    }
}

__global__ void k_finalizeL(const int* __restrict__ Lp, int* __restrict__ meta) {
    int L = *Lp;
    int effW = MS + L; if (effW > MTOT) effW = MTOT;
    meta[0] = effW;
    meta[1] = (effW + 15) / 16;   // column tiles for GEMM2
}

// ---------------- build concatenated normalized features scN[b][w][c][NCAP] ----------------
__global__ void k_scn(const float* __restrict__ supN, const float* __restrict__ unlN,
                      const int* __restrict__ compidx, const int* __restrict__ counts,
                      float* __restrict__ scn) {
    int idx = blockIdx.x * blockDim.x + threadIdx.x;
    if (idx >= B_ * NW * NCAP) return;
    int n = idx % NCAP, bw = idx / NCAP;
    int b = bw / NW, w = bw % NW;
    float* dst = scn + ((size_t)bw * C_) * NCAP + n;
    if (n < MS) {
        const float* s = supN + (size_t)b * C_ * SUPC + (w * MS + n);
        for (int c = 0; c < C_; ++c) dst[(size_t)c * NCAP] = s[(size_t)c * SUPC];
    } else {
        int j = n - MS;
        int cnt = counts[b * NW + w];
        if (j < cnt) {
            int m = compidx[(size_t)bw * MU + j];
            const float* s = unlN + (size_t)b * C_ * MU + m;
            for (int c = 0; c < C_; ++c) dst[(size_t)c * NCAP] = s[(size_t)c * MU];
        } else {
            for (int c = 0; c < C_; ++c) dst[(size_t)c * NCAP] = 0.f;  // zero rows -> sim 0.5
        }
    }
}

// ---------------- fused GEMM2 + mutual-NN + CE loss, one block per (b,q) ----------------
__global__ void __launch_bounds__(256) k_final(const float* __restrict__ qN,
                                               const float* __restrict__ scn,
                                               const int* __restrict__ qy,
                                               const int* __restrict__ meta,
                                               float* __restrict__ out) {
    __shared__ float qs[C_ * 32];                 // A-panel: [c][mq] (mq padded to 32)
    __shared__ unsigned colmaxU[NW * MQ];         // max over n, float bits (vals >= 0)
    __shared__ unsigned long long bestkey[MQ];    // (valbits<<32) | ~(w*MTOT+n)
    __shared__ int gArr[MQ];
    __shared__ float qv[NW];
    int b = blockIdx.x / Q_, q = blockIdx.x % Q_;
    int tid = threadIdx.x;

    const float* qsrc = qN + (size_t)(b * Q_ + q) * C_ * 32;
    for (int i = tid; i < C_ * 32; i += 256) qs[i] = qsrc[i];
    if (tid < NW * MQ) colmaxU[tid] = 0u;
    if (tid < MQ) bestkey[tid] = 0ull;
    if (tid < NW) qv[tid] = 0.f;
    __syncthreads();

    int effW = meta[0], ntiles = meta[1];
    int wid = tid >> 5, lane = tid & 31, hi = lane >> 4, lo = lane & 15;
    int jobs = NW * ntiles * 2;
    for (int job = wid; job < jobs; job += 8) {
        int mt = job & 1, rest = job >> 1;
        int w = rest % NW, nt = rest / NW;
        int n0 = nt * 16;
        const float* Bb = scn + ((size_t)(b * NW + w) * C_) * NCAP + n0 + lo;
        v8f acc = {};
        for (int k0 = 0; k0 < C_; k0 += 4) {
            int ka = k0 + 2 * hi;
            v2f a, bv;
            a[0]  = qs[ka * 32 + mt * 16 + lo];
            a[1]  = qs[(ka + 1) * 32 + mt * 16 + lo];
            bv[0] = Bb[(size_t)ka * NCAP];
            bv[1] = Bb[(size_t)(ka + 1) * NCAP];
            acc = __builtin_amdgcn_wmma_f32_16x16x4_f32(false, a, false, bv,
                                                        (short)0, acc, false, false);
        }
        for (int r = 0; r < 8; ++r) {
            int row = mt * 16 + r + 8 * hi;
            int col = n0 + lo;
            if (row < MQ && col < effW) {
                float val = (acc[r] + 1.0f) * 0.5f;
                unsigned vb = __float_as_uint(val);
                atomicMax(&colmaxU[w * MQ + row], vb);
                unsigned long long key =
                    ((unsigned long long)vb << 32) |
                    (unsigned)(0xFFFFFFFFu - (unsigned)(w * MTOT + col));
                atomicMax(&bestkey[row], key);
            }
        }
    }
    __syncthreads();

    // g[mq] = support_nearest[query_nearest[mq]] (recompute 25-dot column on the fly)
    if (tid < MQ) {
        unsigned ji = 0xFFFFFFFFu - (unsigned)(bestkey[tid] & 0xFFFFFFFFull);
        int wsel = ji / MTOT, nsel = ji % MTOT;
        const float* Bb = scn + ((size_t)(b * NW + wsel) * C_) * NCAP + nsel;
        float best = -3.0e38f; int bi = 0;
        for (int mq2 = 0; mq2 < MQ; ++mq2) {
            float dot = 0.f;
            for (int c = 0; c < C_; ++c) dot += qs[c * 32 + mq2] * Bb[(size_t)c * NCAP];
            if (dot > best) { best = dot; bi = mq2; }   // rescale is monotone
        }
        gArr[tid] = bi;
    }
    __syncthreads();

    if (tid < NW) {
        float s = 0.f;
        for (int mq = 0; mq < MQ; ++mq)
            if (gArr[mq] == mq) s += __uint_as_float(colmaxU[tid * MQ + mq]);
        qv[tid] = s;
    }
    __syncthreads();

    if (tid == 0) {
        int y = qy[b * Q_ + q];
        float mx = qv[0];
        for (int w = 1; w < NW; ++w) mx = fmaxf(mx, qv[w]);
        float se = 0.f;
        for (int w = 0; w < NW; ++w) se += expf(qv[w] - mx);
        float loss = (logf(se) + mx) - qv[y];
        atomicAdd(out, loss / (float)(B_ * Q_));
    }
}

// ---------------- host side ----------------
extern "C" void kernel_launch(void* const* d_in, const int* in_sizes, int n_in,
                              void* d_out, int out_size, void* d_ws, size_t ws_size,
                              hipStream_t stream) {
    const float* sup = (const float*)d_in[0];   // [4,25,64,5,5]
    const float* qx  = (const float*)d_in[2];   // [4,75,64,5,5]
    const int*   qy  = (const int*)d_in[3];     // [4,75]
    const float* unl = (const float*)d_in[4];   // [4,100,64,5,5]
    float* out = (float*)d_out;

    char* p = (char*)d_ws;
    float* unlN    = (float*)p; p += (size_t)B_ * C_ * MU * 4;          // 2.56 MB
    float* supN    = (float*)p; p += (size_t)B_ * C_ * SUPC * 4;        // 0.66 MB
    float* qN      = (float*)p; p += (size_t)B_ * Q_ * C_ * 32 * 4;     // 2.46 MB
    float* pool    = (float*)p; p += (size_t)B_ * MU * SUPC * 4;        // 25.6 MB
    float* scn     = (float*)p; p += (size_t)B_ * NW * C_ * NCAP * 4;   // 13.5 MB
    int* s_nearest = (int*)p;   p += (size_t)B_ * NW * MS * 4;
    int* u_nearest = (int*)p;   p += (size_t)B_ * MU * 4;
    int* cls       = (int*)p;   p += (size_t)B_ * MU * 4;
    int* compidx   = (int*)p;   p += (size_t)B_ * NW * MU * 4;
    int* counts    = (int*)p;   p += (size_t)B_ * NW * 4;
    int* Lp        = (int*)p;   p += 4;
    int* meta      = (int*)p;   p += 8;
    (void)in_sizes; (void)n_in; (void)out_size; (void)ws_size;

    k_init<<<1, 32, 0, stream>>>(out, Lp);
    k_norm_unl<<<(B_ * MU + 255) / 256, 256, 0, stream>>>(unl, unlN);
    k_norm_sup<<<(B_ * SUPC + 255) / 256, 256, 0, stream>>>(sup, supN);
    k_norm_q<<<(B_ * Q_ * 32 + 255) / 256, 256, 0, stream>>>(qx, qN);
    k_gemm1<<<dim3(SUPC / 16, (MU + 15) / 16, B_), 32, 0, stream>>>(unlN, supN, pool);
    k_unearest<<<(B_ * MU + 255) / 256, 256, 0, stream>>>(pool, u_nearest, cls);
    k_snearest<<<B_ * NW * MS, 256, 0, stream>>>(pool, s_nearest);
    k_compact<<<1, B_, 0, stream>>>(u_nearest, s_nearest, cls, compidx, counts, Lp);
    k_finalizeL<<<1, 1, 0, stream>>>(Lp, meta);
    k_scn<<<(B_ * NW * NCAP + 255) / 256, 256, 0, stream>>>(supN, unlN, compidx, counts, scn);
    k_final<<<B_ * Q_, 256, 0, stream>>>(qN, scn, qy, meta, out);
}